// HeteroGCNLayer_85152021611242
// MI455X (gfx1250) — compile-verified
//
#include <hip/hip_runtime.h>
#include <hip/hip_bf16.h>

typedef __attribute__((ext_vector_type(2))) float v2f;
typedef __attribute__((ext_vector_type(8))) float v8f;

#define FDIM 128
#define GEMM_ROWS 64  // rows per block; 4 row-tiles of 16 per wave

// ---------------------------------------------------------------------------
// Zero-fill workspace (degree arrays + aggregation buffers must start at 0)
// ---------------------------------------------------------------------------
__global__ void hgcn_fill_zero(float* __restrict__ p, size_t cnt4) {
  size_t i = blockIdx.x * (size_t)blockDim.x + threadIdx.x;
  size_t stride = (size_t)gridDim.x * blockDim.x;
  float4* p4 = (float4*)p;
  float4 z = make_float4(0.f, 0.f, 0.f, 0.f);
  for (; i < cnt4; i += stride) p4[i] = z;
}

// ---------------------------------------------------------------------------
// Degree accumulation: out_deg over src, in_deg over dst, both relations
// ---------------------------------------------------------------------------
__global__ void hgcn_degree(const int* __restrict__ src_lt, const int* __restrict__ dst_lt,
                            const int* __restrict__ src_do, const int* __restrict__ dst_do,
                            float* __restrict__ outdeg_lt, float* __restrict__ indeg_lt,
                            float* __restrict__ outdeg_do, float* __restrict__ indeg_do,
                            int E) {
  int e = blockIdx.x * blockDim.x + threadIdx.x;
  if (e >= E) return;
  unsafeAtomicAdd(&outdeg_lt[src_lt[e]], 1.0f);
  unsafeAtomicAdd(&indeg_lt[dst_lt[e]], 1.0f);
  unsafeAtomicAdd(&outdeg_do[src_do[e]], 1.0f);
  unsafeAtomicAdd(&indeg_do[dst_do[e]], 1.0f);
}

// ---------------------------------------------------------------------------
// Edge scatter: agg[dst] += x[src] * rsqrt(max(out_deg[src],1))
// One wave (32 lanes) per edge, 4 floats per lane (float4 coalesced gather).
// e is wave-uniform -> scalarize indices with readfirstlane.
// ---------------------------------------------------------------------------
__global__ void hgcn_scatter(const float* __restrict__ x, const float* __restrict__ outdeg,
                             const int* __restrict__ src, const int* __restrict__ dst,
                             float* __restrict__ agg, int E) {
  int t = blockIdx.x * blockDim.x + threadIdx.x;
  int e = t >> 5;  // wave-uniform (wave32, contiguous lanes)
  if (e >= E) return;
  int l = t & 31;
  int s = __builtin_amdgcn_readfirstlane(src[e]);
  int d = __builtin_amdgcn_readfirstlane(dst[e]);
  float sc = rsqrtf(fmaxf(outdeg[s], 1.0f));
  float4 v = ((const float4*)x)[(size_t)s * 32 + l];
  float* dp = agg + (size_t)d * FDIM + l * 4;
  unsafeAtomicAdd(dp + 0, v.x * sc);
  unsafeAtomicAdd(dp + 1, v.y * sc);
  unsafeAtomicAdd(dp + 2, v.z * sc);
  unsafeAtomicAdd(dp + 3, v.w * sc);
}

// ---------------------------------------------------------------------------
// Fused GEMM: out = (agg * rsqrt(max(indeg,1))) @ Wrel + brel + x @ Wself
// Block: 256 threads (8 waves), 64 output rows. Wave w owns cols 16w..16w+15
// and 4 row-tiles -> 4 v8f accumulators; each B fragment feeds 4 WMMAs.
// A fragments from LDS (16x4 f32 A layout), B fragments from global weights.
// ---------------------------------------------------------------------------
__global__ __launch_bounds__(256) void hgcn_gemm(
    const float* __restrict__ agg, const float* __restrict__ indeg,
    const float* __restrict__ x,
    const float* __restrict__ Wrel, const float* __restrict__ brel,
    const float* __restrict__ Wself,
    float* __restrict__ out, int n) {
  __shared__ float sA[GEMM_ROWS * FDIM];
  __shared__ float sX[GEMM_ROWS * FDIM];

  const int row0 = blockIdx.x * GEMM_ROWS;
  const int tid = threadIdx.x;

  // Stage 64x128 tiles of (dst-normalized agg) and x into LDS.
  {
    const float4* aggv = (const float4*)agg;
    const float4* xv = (const float4*)x;
    float4* sA4 = (float4*)sA;
    float4* sX4 = (float4*)sX;
#pragma unroll
    for (int i = 0; i < GEMM_ROWS / 8; ++i) {  // 8 iters: 2048 float4 per array
      int lin = tid + i * 256;  // 0..2047
      int rowl = lin >> 5;      // 0..63
      int c4 = lin & 31;        // float4 column
      int g = row0 + rowl;
      if (g >= n) g = n - 1;    // clamp (safe duplicate read; stores are guarded)
      float sc = rsqrtf(fmaxf(indeg[g], 1.0f));
      float4 av = aggv[(size_t)g * 32 + c4];
      av.x *= sc; av.y *= sc; av.z *= sc; av.w *= sc;
      sA4[rowl * 32 + c4] = av;
      sX4[rowl * 32 + c4] = xv[(size_t)g * 32 + c4];
    }
  }
  __syncthreads();

  const int lane = tid & 31;
  const int wave = tid >> 5;
  const int r = lane & 15;            // row-in-tile (A) / col-in-tile (B)
  const int koff = (lane >> 4) << 1;  // lanes 0-15: K+0,1 ; lanes 16-31: K+2,3
  const int col = (wave << 4) + r;    // output column 0..127

  const float* wr = Wrel + col;
  const float* wsf = Wself + col;
  const float2* sA2 = (const float2*)sA;
  const float2* sX2 = (const float2*)sX;

  v8f acc[4] = {{}, {}, {}, {}};

#pragma unroll 2
  for (int k0 = 0; k0 < FDIM; k0 += 4) {
    const int kk = k0 + koff;
    // B fragments (4x16 f32): lanes 0-15 K=k0..k0+1, lanes 16-31 K=k0+2..k0+3
    v2f b0; b0[0] = wr[kk * FDIM];  b0[1] = wr[(kk + 1) * FDIM];
    v2f bs; bs[0] = wsf[kk * FDIM]; bs[1] = wsf[(kk + 1) * FDIM];
#pragma unroll
    for (int rt = 0; rt < 4; ++rt) {
      // A fragments (16x4 f32) from LDS, 8-byte loads
      float2 a = sA2[((rt * 16 + r) * FDIM + kk) >> 1];
      float2 xx = sX2[((rt * 16 + r) * FDIM + kk) >> 1];
      v2f va; va[0] = a.x;  va[1] = a.y;
      v2f vx; vx[0] = xx.x; vx[1] = xx.y;
      acc[rt] = __builtin_amdgcn_wmma_f32_16x16x4_f32(false, va, false, b0, (short)0,
                                                      acc[rt], false, false);
      acc[rt] = __builtin_amdgcn_wmma_f32_16x16x4_f32(false, vx, false, bs, (short)0,
                                                      acc[rt], false, false);
    }
  }

  // C/D layout: VGPR j -> M=j (lanes 0-15) / M=j+8 (lanes 16-31)
  const float bias = brel[col];
  const int mbase = (lane >> 4) << 3;
  if (row0 + GEMM_ROWS <= n) {
    // Fast path: full block, unguarded coalesced stores.
#pragma unroll
    for (int rt = 0; rt < 4; ++rt) {
#pragma unroll
      for (int j = 0; j < 8; ++j) {
        int row = row0 + rt * 16 + mbase + j;
        out[(size_t)row * FDIM + col] = acc[rt][j] + bias;
      }
    }
  } else {
#pragma unroll
    for (int rt = 0; rt < 4; ++rt) {
#pragma unroll
      for (int j = 0; j < 8; ++j) {
        int row = row0 + rt * 16 + mbase + j;
        if (row < n) out[(size_t)row * FDIM + col] = acc[rt][j] + bias;
      }
    }
  }
}

// ---------------------------------------------------------------------------
extern "C" void kernel_launch(void* const* d_in, const int* in_sizes, int n_in,
                              void* d_out, int out_size, void* d_ws, size_t ws_size,
                              hipStream_t stream) {
  const float* x_ep = (const float*)d_in[0];
  const float* x_flow = (const float*)d_in[1];
  const float* W_lt = (const float*)d_in[2];
  const float* b_lt = (const float*)d_in[3];
  const float* W_do = (const float*)d_in[4];
  const float* b_do = (const float*)d_in[5];
  const float* W_self = (const float*)d_in[6];
  const int* src_lt = (const int*)d_in[7];
  const int* dst_lt = (const int*)d_in[8];
  const int* src_do = (const int*)d_in[9];
  const int* dst_do = (const int*)d_in[10];

  const int n_ep = in_sizes[0] / FDIM;
  const int n_flow = in_sizes[1] / FDIM;
  const int E = in_sizes[7];

  float* out_flow = (float*)d_out;
  float* out_ep = out_flow + (size_t)n_flow * FDIM;

  // Workspace layout
  float* ws = (float*)d_ws;
  float* agg_flow = ws;                              // n_flow*128
  float* agg_ep = agg_flow + (size_t)n_flow * FDIM;  // n_ep*128
  float* outdeg_lt = agg_ep + (size_t)n_ep * FDIM;   // n_ep
  float* indeg_lt = outdeg_lt + n_ep;                // n_flow
  float* outdeg_do = indeg_lt + n_flow;              // n_flow
  float* indeg_do = outdeg_do + n_flow;              // n_ep

  size_t total_floats = (size_t)n_flow * FDIM + (size_t)n_ep * FDIM +
                        2 * ((size_t)n_ep + (size_t)n_flow);
  size_t cnt4 = total_floats / 4;  // all components are multiples of 4

  hgcn_fill_zero<<<4096, 256, 0, stream>>>(ws, cnt4);

  hgcn_degree<<<(E + 255) / 256, 256, 0, stream>>>(
      src_lt, dst_lt, src_do, dst_do,
      outdeg_lt, indeg_lt, outdeg_do, indeg_do, E);

  // relation endpoint -links_to-> flow
  {
    long long threads = (long long)E * 32;
    hgcn_scatter<<<(unsigned)((threads + 255) / 256), 256, 0, stream>>>(
        x_ep, outdeg_lt, src_lt, dst_lt, agg_flow, E);
  }
  // relation flow -depends_on-> endpoint
  {
    long long threads = (long long)E * 32;
    hgcn_scatter<<<(unsigned)((threads + 255) / 256), 256, 0, stream>>>(
        x_flow, outdeg_do, src_do, dst_do, agg_ep, E);
  }

  hgcn_gemm<<<(n_flow + GEMM_ROWS - 1) / GEMM_ROWS, 256, 0, stream>>>(
      agg_flow, indeg_lt, x_flow, W_lt, b_lt, W_self, out_flow, n_flow);
  hgcn_gemm<<<(n_ep + GEMM_ROWS - 1) / GEMM_ROWS, 256, 0, stream>>>(
      agg_ep, indeg_do, x_ep, W_do, b_do, W_self, out_ep, n_ep);
}